// Rewirescorelayer_14800457302374
// MI455X (gfx1250) — compile-verified
//
#include <hip/hip_runtime.h>
#include <hip/hip_bf16.h>
#include <math.h>

// Problem constants (match reference)
#define R_     2
#define N_     4096
#define FIN_   256
#define D_     64
#define H_     8
#define W_     128
#define K_     16
#define WC_    32          // N_/W_
#define DH_    512         // D_*H_
#define BIGN_  8192        // R_*N_

typedef __attribute__((ext_vector_type(2))) float v2f;
typedef __attribute__((ext_vector_type(4))) float v4f;
typedef __attribute__((ext_vector_type(8))) float v8f;

// ---------------------------------------------------------------------------
// Kernel 1: zero the 8192x8192 fp32 output (the bandwidth-bound part).
// Output (268 MB) exceeds the 192 MB L2 and is never re-read on device, so
// use non-temporal (NT) stores to stream past L2 instead of thrashing it.
// ---------------------------------------------------------------------------
__global__ __launch_bounds__(256) void zero_out_kernel(v4f* __restrict__ p,
                                                       long n4) {
  long i = (long)blockIdx.x * blockDim.x + threadIdx.x;
  long stride = (long)gridDim.x * blockDim.x;
  v4f z = (v4f){0.f, 0.f, 0.f, 0.f};
  for (; i < n4; i += stride) __builtin_nontemporal_store(z, &p[i]);
}

// ---------------------------------------------------------------------------
// Kernel 2: fused Q/K projection GEMM with V_WMMA_F32_16X16X4_F32.
// One wave per 16x16 tile of the [4096 x 512] output (per relation); the
// A-fragment (node features) is reused for both the Q and K accumulators.
//
// A 16x4 layout: lanes 0-15 hold M=lane; VGPR0 = K{0 | 2}, VGPR1 = K{1 | 3}
// (second K per half-wave).  B 4x16 mirrors it with N=lane&15.
// C/D: VGPR v -> M = v (lanes 0-15) / v+8 (lanes 16-31), N = lane&15.
// ---------------------------------------------------------------------------
__global__ __launch_bounds__(256) void proj_qk_kernel(
    const float* __restrict__ X,   // [R, N, FIN]
    const float* __restrict__ Wq,  // [FIN, DH]
    const float* __restrict__ bq,  // [DH]
    const float* __restrict__ Wk,  // [FIN, DH]
    const float* __restrict__ bk,  // [DH]
    float* __restrict__ Qo,        // [R, N, DH]
    float* __restrict__ Ko)        // [R, N, DH]
{
  const int wave = (blockIdx.x * blockDim.x + threadIdx.x) >> 5;  // 0..16383
  const int lane = threadIdx.x & 31;
  const int nt = wave & 31;           // 32 N-tiles of 16 (DH=512)
  const int mt = (wave >> 5) & 255;   // 256 M-tiles of 16 (N=4096)
  const int r  = wave >> 13;          // relation

  const int l15  = lane & 15;
  const int koff = (lane >> 4) << 1;  // 0 or 2: K pair per half-wave
  const int m = mt * 16 + l15;
  const int n = nt * 16 + l15;

  const float* xrow = X + ((size_t)r * N_ + m) * FIN_;

  v8f accQ = {}; v8f accK = {};
  for (int kk = 0; kk < FIN_; kk += 4) {
    const int k0 = kk + koff;
    v2f a = *(const v2f*)(xrow + k0);            // X[m][k0], X[m][k0+1]
    v2f bQ, bK;
    bQ.x = Wq[(size_t)k0 * DH_ + n];
    bQ.y = Wq[(size_t)(k0 + 1) * DH_ + n];
    bK.x = Wk[(size_t)k0 * DH_ + n];
    bK.y = Wk[(size_t)(k0 + 1) * DH_ + n];
    accQ = __builtin_amdgcn_wmma_f32_16x16x4_f32(false, a, false, bQ,
                                                 (short)0, accQ, false, false);
    accK = __builtin_amdgcn_wmma_f32_16x16x4_f32(false, a, false, bK,
                                                 (short)0, accK, false, false);
  }

  const float biasQ = bq[n];
  const float biasK = bk[n];
  const int mbase = mt * 16 + ((lane >> 4) << 3);  // +0 or +8 per half-wave
  #pragma unroll
  for (int v = 0; v < 8; ++v) {
    const size_t row = (size_t)r * N_ + mbase + v;
    Qo[row * DH_ + n] = accQ[v] + biasQ;
    Ko[row * DH_ + n] = accK[v] + biasK;
  }
}

// ---------------------------------------------------------------------------
// Kernel 3: per-(relation, window) block:
//   scores_h = (Q_w K_w^T) * 8  -> softmax rows -> mean over heads  (WMMA +
//   in-register softmax via shfl_xor), then per-row gumbel softmax + top-16
//   straight-through, scattered into the output diagonal block.
// ---------------------------------------------------------------------------
#define LKS 66    // LDS stride for K tile (2j+k0 mod 64 distinct -> no conflicts)
#define LZS 130   // LDS stride for z buffer
#define LAS 129   // LDS stride for attn buffer

__global__ __launch_bounds__(256) void attn_topk_kernel(
    const float* __restrict__ Q,   // [R, N, DH]
    const float* __restrict__ Kv,  // [R, N, DH]
    const float* __restrict__ GU,  // [R, WC, W, W] uniform noise
    float* __restrict__ out)       // [BIGN, BIGN]
{
  __shared__ float ldsKZ[W_ * LZS];  // head-K staging (stride LKS) / z buffer
  __shared__ float ldsA[W_ * LAS];   // accumulated mean attention

  const int tid  = threadIdx.x;
  const int lane = tid & 31;
  const int w    = tid >> 5;        // wave id 0..7 -> row strip
  const int blk  = blockIdx.x;      // 0..63
  const int r    = blk >> 5;
  const int wc   = blk & 31;

  const int l15   = lane & 15;
  const int koff  = (lane >> 4) << 1;
  const int half8 = (lane >> 4) << 3;

  const size_t rowBase = (size_t)r * N_ + (size_t)wc * W_;

  v8f attnAcc[8];
  #pragma unroll
  for (int i = 0; i < 8; ++i) attnAcc[i] = (v8f){};

  for (int h = 0; h < H_; ++h) {
    __syncthreads();  // previous head's compute done before restaging
    // Stage K_w,h [128 x 64] into LDS (float2 chunks, coalesced).
    for (int i = tid; i < W_ * 32; i += 256) {
      const int j  = i >> 5;
      const int dp = (i & 31) << 1;
      const float* src = Kv + (rowBase + j) * DH_ + h * D_ + dp;
      *(v2f*)&ldsKZ[j * LKS + dp] = *(const v2f*)src;
    }
    __syncthreads();

    // Wave w computes rows [16w, 16w+16) x all 128 cols: 8 tiles x 16 K-steps.
    v8f acc[8];
    #pragma unroll
    for (int nt = 0; nt < 8; ++nt) acc[nt] = (v8f){};
    const float* qrow = Q + (rowBase + w * 16 + l15) * DH_ + h * D_;
    for (int kk = 0; kk < D_; kk += 4) {
      const int k0 = kk + koff;
      v2f a = *(const v2f*)(qrow + k0);
      #pragma unroll
      for (int nt = 0; nt < 8; ++nt) {
        v2f b = *(const v2f*)&ldsKZ[(nt * 16 + l15) * LKS + k0];
        acc[nt] = __builtin_amdgcn_wmma_f32_16x16x4_f32(
            false, a, false, b, (short)0, acc[nt], false, false);
      }
    }

    // In-register row softmax. Row for component v lives in 16-lane halves;
    // xor masks 1,2,4,8 stay within each half (wave32).
    #pragma unroll
    for (int v = 0; v < 8; ++v) {
      float mx = -3.4e38f;
      #pragma unroll
      for (int nt = 0; nt < 8; ++nt) mx = fmaxf(mx, acc[nt][v]);
      for (int d = 1; d < 16; d <<= 1) mx = fmaxf(mx, __shfl_xor(mx, d, 32));
      float e[8]; float sum = 0.f;
      #pragma unroll
      for (int nt = 0; nt < 8; ++nt) {
        e[nt] = expf((acc[nt][v] - mx) * 8.0f);  // scores = raw * sqrt(D)=8
        sum += e[nt];
      }
      for (int d = 1; d < 16; d <<= 1) sum += __shfl_xor(sum, d, 32);
      const float inv = 0.125f / sum;            // head mean (1/H)
      #pragma unroll
      for (int nt = 0; nt < 8; ++nt) attnAcc[nt][v] += e[nt] * inv;
    }
  }

  // Dump mean attention to LDS (padded stride).
  __syncthreads();
  #pragma unroll
  for (int v = 0; v < 8; ++v) {
    const int row = w * 16 + half8 + v;
    #pragma unroll
    for (int nt = 0; nt < 8; ++nt)
      ldsA[row * LAS + nt * 16 + l15] = attnAcc[nt][v];
  }
  __syncthreads();

  // Phase 2: one thread per row -> gumbel softmax + top-16 straight-through.
  if (tid < W_) {
    const int row = tid;
    const float* gu = GU + (((size_t)r * WC_ + wc) * W_ + row) * W_;
    float* zrow = &ldsKZ[row * LZS];

    float zmax = -3.4e38f;
    for (int j = 0; j < W_; ++j) {
      const float g = -logf(-logf(gu[j]));
      const float z = (ldsA[row * LAS + j] + g) * 2.0f;  // 1/TAU = 2
      zrow[j] = z;
      zmax = fmaxf(zmax, z);
    }
    float zsum = 0.f;
    for (int j = 0; j < W_; ++j) {
      const float e = expf(zrow[j] - zmax);
      zrow[j] = e;
      zsum += e;
    }
    const float invz = 1.0f / zsum;

    // top-16 on clean attention logits (ties -> lowest index, like top_k)
    unsigned msk[4] = {0u, 0u, 0u, 0u};
    for (int it = 0; it < K_; ++it) {
      float best = -3.4e38f; int bi = 0;
      for (int j = 0; j < W_; ++j) {
        if (!((msk[j >> 5] >> (j & 31)) & 1u)) {
          const float a = ldsA[row * LAS + j];
          if (a > best) { best = a; bi = j; }
        }
      }
      msk[bi >> 5] |= 1u << (bi & 31);
    }

    // Stream the diagonal-block rows past L2 as well (write-once data).
    float* orow = out + (rowBase + row) * (size_t)BIGN_ + rowBase;
    for (int j = 0; j < W_; ++j) {
      const float s = zrow[j] * invz;
      const bool top = (msk[j >> 5] >> (j & 31)) & 1u;
      float y = top ? ((1.0f - s) + s) : 0.0f;  // straight-through value
      y = fminf(fmaxf(y, 0.0f), 1.0f);
      __builtin_nontemporal_store(y, &orow[j]);
    }
  }
}

// ---------------------------------------------------------------------------
// Launch: zero-fill -> projections -> window attention/top-k.
// Workspace: Q [R*N*DH] + K [R*N*DH] fp32 = 32 MB.
// ---------------------------------------------------------------------------
extern "C" void kernel_launch(void* const* d_in, const int* in_sizes, int n_in,
                              void* d_out, int out_size, void* d_ws,
                              size_t ws_size, hipStream_t stream) {
  const float* X  = (const float*)d_in[0];
  const float* Wq = (const float*)d_in[1];
  const float* bq = (const float*)d_in[2];
  const float* Wk = (const float*)d_in[3];
  const float* bk = (const float*)d_in[4];
  const float* gu = (const float*)d_in[5];
  float* out = (float*)d_out;

  float* Qws = (float*)d_ws;                       // [R*N*DH]
  float* Kws = Qws + (size_t)R_ * N_ * DH_;        // [R*N*DH]

  const long n4 = (long)BIGN_ * BIGN_ / 4;
  zero_out_kernel<<<8192, 256, 0, stream>>>((v4f*)out, n4);

  // 16384 tiles / 8 waves per block = 2048 blocks
  proj_qk_kernel<<<2048, 256, 0, stream>>>(X, Wq, bq, Wk, bk, Qws, Kws);

  attn_topk_kernel<<<R_ * WC_, 256, 0, stream>>>(Qws, Kws, gu, out);
}